// ChebLayer_66632122630757
// MI455X (gfx1250) — compile-verified
//
#include <hip/hip_runtime.h>
#include <hip/hip_bf16.h>

typedef __attribute__((ext_vector_type(2))) float v2f;
typedef __attribute__((ext_vector_type(8))) float v8f;

#define N_DIM 2048
#define F_DIM 128
#define B_DIM 8
#define BM 128
#define BK 32
#define A_LD 36   // A tile row stride (floats): even, mult of 4 -> aligned b64/b128, no bank conflicts
#define B_LD 288  // B tile pair-row stride (floats): == 32 mod 64 -> half-wave bank split

// ---------------------------------------------------------------------------
// Kernel 1: d_inv_sqrt[b*N+i] = rsqrt(sum_j adj[b,i,j] + 1e-6)
// ---------------------------------------------------------------------------
__global__ __launch_bounds__(256) void cheb_deg_kernel(
    const float* __restrict__ adj, float* __restrict__ dis) {
  const int row  = blockIdx.x * 8 + (threadIdx.x >> 5);
  const int lane = threadIdx.x & 31;
  const float4* arow = (const float4*)(adj + (size_t)row * N_DIM);
  float s = 0.0f;
  #pragma unroll 4
  for (int j = lane; j < N_DIM / 4; j += 32) {
    float4 v = arow[j];
    s += (v.x + v.y) + (v.z + v.w);
  }
  #pragma unroll
  for (int off = 16; off > 0; off >>= 1) s += __shfl_xor(s, off, 32);
  if (lane == 0) dis[row] = rsqrtf(s + 1e-6f);
}

// ---------------------------------------------------------------------------
// Kernel 2/3: Y[b] = alpha * (L[b] @ X[b]) + beta * Z[b]
// L formed on the fly; register-prefetch + LDS ping-pong pipeline;
// fp32 WMMA 16x16x4, wave owns a 16x128 strip (8 x v8f accumulators).
// ---------------------------------------------------------------------------
__global__ __launch_bounds__(256) void chebL_spmm_kernel(
    const float* __restrict__ adj, const float* __restrict__ dis,
    const float* __restrict__ X, const float* __restrict__ Z,
    float* __restrict__ Y, float alpha, float beta) {
  const int b  = blockIdx.y;
  const int m0 = blockIdx.x * BM;
  const float* adjB = adj + (size_t)b * N_DIM * N_DIM;
  const float* disB = dis + (size_t)b * N_DIM;
  const float* XB   = X   + (size_t)b * N_DIM * F_DIM;
  float*       YB   = Y   + (size_t)b * N_DIM * F_DIM;

  __shared__ float As[2][BM * A_LD];        // L tile, row-major, stride 36
  __shared__ float Bs[2][(BK / 2) * B_LD];  // X tile, k-pair interleaved

  const int tid  = threadIdx.x;
  const int lane = tid & 31;
  const int wave = tid >> 5;

  v8f acc[8] = {};

  // A staging: thread -> (row = tid>>3 (+32r), col4 = (tid&7)*4)
  const int arow = tid >> 3;
  const int ac4  = (tid & 7) * 4;
  // B staging: thread -> (pair = tid>>5 (+8r), col4 = (tid&31)*4)
  const int bp  = tid >> 5;
  const int bn4 = (tid & 31) * 4;

  // Fragment coordinates (ISA 7.12.2)
  const int mrow = wave * 16 + (lane & 15);
  const int g2   = (lane >> 4) * 2;  // K half select
  const int grp  = lane >> 4;
  const int ncol = lane & 15;

  float4 avr[4]; float dvr[4]; float4 djr;
  float4 bxa[2], bxb[2];

  auto fetch = [&](int k0) {
    djr = *(const float4*)&disB[k0 + ac4];
    #pragma unroll
    for (int r = 0; r < 4; ++r) {
      const int gi = m0 + arow + r * 32;
      avr[r] = *(const float4*)&adjB[(size_t)gi * N_DIM + k0 + ac4];
      dvr[r] = disB[gi];
    }
    #pragma unroll
    for (int r = 0; r < 2; ++r) {
      const int kk = (bp + r * 8) * 2;
      bxa[r] = *(const float4*)&XB[(size_t)(k0 + kk) * F_DIM + bn4];
      bxb[r] = *(const float4*)&XB[(size_t)(k0 + kk + 1) * F_DIM + bn4];
    }
  };

  auto commit = [&](int k0, int buf) {
    #pragma unroll
    for (int r = 0; r < 4; ++r) {
      const int i  = arow + r * 32;
      const int gi = m0 + i;
      const int gj = k0 + ac4;
      float4 v;
      v.x = -dvr[r] * avr[r].x * djr.x + ((gi == gj + 0) ? 1.0f : 0.0f);
      v.y = -dvr[r] * avr[r].y * djr.y + ((gi == gj + 1) ? 1.0f : 0.0f);
      v.z = -dvr[r] * avr[r].z * djr.z + ((gi == gj + 2) ? 1.0f : 0.0f);
      v.w = -dvr[r] * avr[r].w * djr.w + ((gi == gj + 3) ? 1.0f : 0.0f);
      *(float4*)&As[buf][i * A_LD + ac4] = v;
    }
    #pragma unroll
    for (int r = 0; r < 2; ++r) {
      const int pp = bp + r * 8;
      float4 lo = make_float4(bxa[r].x, bxb[r].x, bxa[r].y, bxb[r].y);
      float4 hi = make_float4(bxa[r].z, bxb[r].z, bxa[r].w, bxb[r].w);
      float* dst = &Bs[buf][pp * B_LD + bn4 * 2];
      *(float4*)(dst + 0) = lo;
      *(float4*)(dst + 4) = hi;
    }
  };

  fetch(0);
  commit(0, 0);
  __syncthreads();

  const int NC = N_DIM / BK;  // 64 chunks
  #pragma unroll 1
  for (int c = 0; c < NC; ++c) {
    const int buf = c & 1;
    if (c + 1 < NC) fetch((c + 1) * BK);  // global loads overlap WMMAs below

    const float* Ab = As[buf];
    const float* Bb = Bs[buf];
    #pragma unroll
    for (int ks = 0; ks < BK; ks += 4) {
      const v2f a = *(const v2f*)&Ab[mrow * A_LD + ks + g2];
      #pragma unroll
      for (int nt = 0; nt < 8; ++nt) {
        const v2f bf =
            *(const v2f*)&Bb[(ks / 2 + grp) * B_LD + (nt * 16 + ncol) * 2];
        acc[nt] = __builtin_amdgcn_wmma_f32_16x16x4_f32(
            false, a, false, bf, (short)0, acc[nt], false, false);
      }
    }

    if (c + 1 < NC) commit((c + 1) * BK, buf ^ 1);  // waits land after WMMAs
    __syncthreads();
  }

  // Epilogue: Y = alpha*acc + beta*Z
  const int mbase = m0 + wave * 16 + grp * 8;
  const float* ZB = (beta != 0.0f) ? (Z + (size_t)b * N_DIM * F_DIM) : nullptr;
  #pragma unroll
  for (int nt = 0; nt < 8; ++nt) {
    const int col = nt * 16 + ncol;
    #pragma unroll
    for (int v = 0; v < 8; ++v) {
      const int row = mbase + v;
      float val = alpha * acc[nt][v];
      if (beta != 0.0f) val += beta * ZB[(size_t)row * F_DIM + col];
      YB[(size_t)row * F_DIM + col] = val;
    }
  }
}

// ---------------------------------------------------------------------------
// Kernel 4: out = relu( [Tx0|Tx1|Tx2] (16384x384) @ Wstk (384x128) + bsum )
// Same pipeline; A-tile source segment switches with the k chunk.
// ---------------------------------------------------------------------------
__global__ __launch_bounds__(256) void cheb_out_kernel(
    const float* __restrict__ Tx0, const float* __restrict__ Tx1,
    const float* __restrict__ Tx2, const float* __restrict__ Wm,
    const float* __restrict__ bias, float* __restrict__ out) {
  const int m0 = blockIdx.x * BM;

  __shared__ float As[2][BM * A_LD];
  __shared__ float Bs[2][(BK / 2) * B_LD];

  const int tid  = threadIdx.x;
  const int lane = tid & 31;
  const int wave = tid >> 5;

  v8f acc[8] = {};

  const int arow = tid >> 3;
  const int ac4  = (tid & 7) * 4;
  const int bp   = tid >> 5;
  const int bn4  = (tid & 31) * 4;

  const int mrow = wave * 16 + (lane & 15);
  const int g2   = (lane >> 4) * 2;
  const int grp  = lane >> 4;
  const int ncol = lane & 15;

  float4 avr[4];
  float4 bxa[2], bxb[2];

  auto fetch = [&](int c) {
    const int k0 = c * BK;
    const float* seg = (c < 4) ? Tx0 : ((c < 8) ? Tx1 : Tx2);
    const int kin = k0 & (F_DIM - 1);
    #pragma unroll
    for (int r = 0; r < 4; ++r) {
      const int gi = m0 + arow + r * 32;
      avr[r] = *(const float4*)&seg[(size_t)gi * F_DIM + kin + ac4];
    }
    #pragma unroll
    for (int r = 0; r < 2; ++r) {
      const int kk = (bp + r * 8) * 2;
      bxa[r] = *(const float4*)&Wm[(size_t)(k0 + kk) * F_DIM + bn4];
      bxb[r] = *(const float4*)&Wm[(size_t)(k0 + kk + 1) * F_DIM + bn4];
    }
  };

  auto commit = [&](int buf) {
    #pragma unroll
    for (int r = 0; r < 4; ++r) {
      const int i = arow + r * 32;
      *(float4*)&As[buf][i * A_LD + ac4] = avr[r];
    }
    #pragma unroll
    for (int r = 0; r < 2; ++r) {
      const int pp = bp + r * 8;
      float4 lo = make_float4(bxa[r].x, bxb[r].x, bxa[r].y, bxb[r].y);
      float4 hi = make_float4(bxa[r].z, bxb[r].z, bxa[r].w, bxb[r].w);
      float* dst = &Bs[buf][pp * B_LD + bn4 * 2];
      *(float4*)(dst + 0) = lo;
      *(float4*)(dst + 4) = hi;
    }
  };

  fetch(0);
  commit(0);
  __syncthreads();

  const int NC = (3 * F_DIM) / BK;  // 12 chunks
  #pragma unroll 1
  for (int c = 0; c < NC; ++c) {
    const int buf = c & 1;
    if (c + 1 < NC) fetch(c + 1);

    const float* Ab = As[buf];
    const float* Bb = Bs[buf];
    #pragma unroll
    for (int ks = 0; ks < BK; ks += 4) {
      const v2f a = *(const v2f*)&Ab[mrow * A_LD + ks + g2];
      #pragma unroll
      for (int nt = 0; nt < 8; ++nt) {
        const v2f bf =
            *(const v2f*)&Bb[(ks / 2 + grp) * B_LD + (nt * 16 + ncol) * 2];
        acc[nt] = __builtin_amdgcn_wmma_f32_16x16x4_f32(
            false, a, false, bf, (short)0, acc[nt], false, false);
      }
    }

    if (c + 1 < NC) commit(buf ^ 1);
    __syncthreads();
  }

  const int mbase = m0 + wave * 16 + grp * 8;
  #pragma unroll
  for (int nt = 0; nt < 8; ++nt) {
    const int col = nt * 16 + ncol;
    const float bs = bias[col] + bias[F_DIM + col] + bias[2 * F_DIM + col];
    #pragma unroll
    for (int v = 0; v < 8; ++v) {
      const int row = mbase + v;
      float val = acc[nt][v] + bs;
      out[(size_t)row * F_DIM + col] = val > 0.0f ? val : 0.0f;
    }
  }
}

// ---------------------------------------------------------------------------
extern "C" void kernel_launch(void* const* d_in, const int* in_sizes, int n_in,
                              void* d_out, int out_size, void* d_ws, size_t ws_size,
                              hipStream_t stream) {
  const float* x   = (const float*)d_in[0]; // [8,2048,128]
  const float* adj = (const float*)d_in[1]; // [8,2048,2048]
  const float* W   = (const float*)d_in[2]; // [3,128,128] == [384,128] stacked
  const float* bb  = (const float*)d_in[3]; // [3,128]
  float* out = (float*)d_out;               // [8,2048,128]

  float* dis = (float*)d_ws;
  float* Tx1 = dis + (size_t)B_DIM * N_DIM;
  float* Tx2 = Tx1 + (size_t)B_DIM * N_DIM * F_DIM;

  cheb_deg_kernel<<<(B_DIM * N_DIM) / 8, 256, 0, stream>>>(adj, dis);

  dim3 grid(N_DIM / BM, B_DIM);
  chebL_spmm_kernel<<<grid, 256, 0, stream>>>(adj, dis, x, nullptr, Tx1,
                                              1.0f, 0.0f);
  chebL_spmm_kernel<<<grid, 256, 0, stream>>>(adj, dis, Tx1, x, Tx2,
                                              2.0f, -1.0f);
  cheb_out_kernel<<<(B_DIM * N_DIM) / BM, 256, 0, stream>>>(x, Tx1, Tx2, W, bb,
                                                            out);
}